// GraphConvolution_60120952209844
// MI455X (gfx1250) — compile-verified
//
#include <hip/hip_runtime.h>

// GCN layer: out = segment_sum((x@W)[edge_col] * edge_weight, edge_row) + bias
// Phase 1: fp32 WMMA GEMM (v_wmma_f32_16x16x4_f32), keeps full f32 precision.
// Phase 2: bias broadcast init.
// Phase 3: per-edge wave32 scatter with hardware f32 atomics (L2-resident).

typedef float v2f __attribute__((ext_vector_type(2)));
typedef float v4f __attribute__((ext_vector_type(4)));
typedef float v8f __attribute__((ext_vector_type(8)));

#define DIN  128
#define DOUT 128

// ---------------------------------------------------------------------------
// Kernel 1: support = x @ W using V_WMMA_F32_16X16X4_F32.
// One wave per 16x16 output tile; K swept in steps of 4 (32 WMMAs per tile).
//
// A (16x4 f32) per-lane layout (ISA 7.12.2):
//   lanes 0-15 : M=lane,    VGPR0=K0, VGPR1=K1
//   lanes 16-31: M=lane-16, VGPR0=K2, VGPR1=K3
// -> lane loads x[row0+(l&15)][k + 2*(l>>4)] as a 2-float (b64) load.
// B (4x16 f32) mirrors A with N=lane&15: two b32 loads from W (stride DOUT).
// C/D (16x16 f32, 8 VGPRs): VGPR v holds row (v + 8*(lane>>4)), col (lane&15).
// ---------------------------------------------------------------------------
__global__ __launch_bounds__(256) void gcn_gemm_wmma(
    const float* __restrict__ x, const float* __restrict__ w,
    float* __restrict__ support, int n_nodes) {
  const int lane = threadIdx.x & 31;
  const int wave = threadIdx.x >> 5;
  const int ntn  = DOUT / 16;                       // 8 tiles along N
  const long ntm = (long)(n_nodes + 15) >> 4;       // tiles along M
  long tile = (long)blockIdx.x * (blockDim.x >> 5) + wave;
  if (tile >= ntm * ntn) return;

  const int  tn   = (int)(tile % ntn);
  const long tm   = tile / ntn;
  const int  row0 = (int)(tm << 4);
  const int  col0 = tn << 4;
  const int  half = lane >> 4;                      // 0 or 1
  const int  l15  = lane & 15;

  // Clamp A row so every lane stays valid: EXEC must be all-1s for WMMA.
  int arow = row0 + l15;
  if (arow >= n_nodes) arow = n_nodes - 1;
  const float* ap = x + (long)arow * DIN + 2 * half;
  const float* bp = w + (long)(2 * half) * DOUT + col0 + l15;

  v8f acc = {};
#pragma unroll
  for (int k = 0; k < DIN; k += 4) {
    v2f a = *(const v2f*)(ap + k);                  // K = k+2*half, k+2*half+1
    v2f b;
    b.x = bp[k * DOUT];                             // W[k+2*half    ][col]
    b.y = bp[k * DOUT + DOUT];                      // W[k+2*half + 1][col]
    // 8 args: (neg_a, A, neg_b, B, c_mod, C, reuse_a, reuse_b)
    acc = __builtin_amdgcn_wmma_f32_16x16x4_f32(
        false, a, false, b, (short)0, acc, false, false);
  }

  float* op = support + (long)row0 * DOUT + col0 + l15;
#pragma unroll
  for (int v = 0; v < 8; ++v) {
    const int r = row0 + v + 8 * half;
    if (r < n_nodes) op[(long)(v + 8 * half) * DOUT] = acc[v];
  }
}

// ---------------------------------------------------------------------------
// Kernel 2: out[i][d] = bias[d]  (float4 stores; DOUT=128 -> 32 float4 / row)
// ---------------------------------------------------------------------------
__global__ __launch_bounds__(256) void gcn_init_bias(
    float* __restrict__ out, const float* __restrict__ bias, long n4) {
  long i = (long)blockIdx.x * blockDim.x + threadIdx.x;
  if (i >= n4) return;
  ((v4f*)out)[i] = ((const v4f*)bias)[i & (DOUT / 4 - 1)];
}

// ---------------------------------------------------------------------------
// Kernel 3: one wave32 per edge.
//   lane loads support[c][lane*4 .. +3] as b128, then 4 hardware f32 atomics
//   into out[r][...]. Both arrays (25.6 MB each) are L2-resident on MI455X.
// ---------------------------------------------------------------------------
__global__ __launch_bounds__(256) void gcn_scatter(
    const float* __restrict__ support, const float* __restrict__ ew,
    const int* __restrict__ erow, const int* __restrict__ ecol,
    float* __restrict__ out, int n_edges) {
  const int lane = threadIdx.x & 31;
  long e = (long)blockIdx.x * (blockDim.x >> 5) + (threadIdx.x >> 5);
  if (e >= n_edges) return;

  const int   r   = erow[e];
  const int   c   = ecol[e];
  const float wgt = ew[e];

  const v4f s = *(const v4f*)(support + (long)c * DOUT + lane * 4);
  float* o = out + (long)r * DOUT + lane * 4;
  unsafeAtomicAdd(o + 0, wgt * s.x);
  unsafeAtomicAdd(o + 1, wgt * s.y);
  unsafeAtomicAdd(o + 2, wgt * s.z);
  unsafeAtomicAdd(o + 3, wgt * s.w);
}

// ---------------------------------------------------------------------------
// Launch
// ---------------------------------------------------------------------------
extern "C" void kernel_launch(void* const* d_in, const int* in_sizes, int n_in,
                              void* d_out, int out_size, void* d_ws, size_t ws_size,
                              hipStream_t stream) {
  const float* x       = (const float*)d_in[0];
  const float* weight  = (const float*)d_in[1];
  const float* bias    = (const float*)d_in[2];
  const float* eweight = (const float*)d_in[3];
  const int*   erow    = (const int*)d_in[4];   // JAX default x64-off -> int32
  const int*   ecol    = (const int*)d_in[5];
  float* out     = (float*)d_out;
  float* support = (float*)d_ws;                // n_nodes*DOUT f32 scratch

  const int n_nodes = in_sizes[0] / DIN;        // 50000
  const int n_edges = in_sizes[3];              // 800000

  // 1) support = x @ W   (8 waves/block, one 16x16 tile per wave)
  const long ntiles  = (long)((n_nodes + 15) / 16) * (DOUT / 16);
  const int  gblocks = (int)((ntiles + 7) / 8);
  gcn_gemm_wmma<<<gblocks, 256, 0, stream>>>(x, weight, support, n_nodes);

  // 2) out = bias (broadcast)
  const long n4      = (long)n_nodes * (DOUT / 4);
  const int  iblocks = (int)((n4 + 255) / 256);
  gcn_init_bias<<<iblocks, 256, 0, stream>>>(out, bias, n4);

  // 3) scatter-add over edges (8 edges per 256-thread block)
  const int sblocks = (n_edges + 7) / 8;
  gcn_scatter<<<sblocks, 256, 0, stream>>>(support, eweight, erow, ecol, out,
                                           n_edges);
}